// UnsupervisedGraphSAGE_84361747628642
// MI455X (gfx1250) — compile-verified
//
#include <hip/hip_runtime.h>
#include <math.h>

// ---------------------------------------------------------------------------
// UnsupervisedGraphSAGE forward on gfx1250 (CDNA5), f32 WMMA path.
// All matmuls use V_WMMA_F32_16X16X4_F32. Each wave computes 4 row-tiles
// (M=64) per 16-col tile so the B fragment is loaded once and reused across
// 4 WMMAs (loads/WMMA: 3 -> 1.5), since the workload is L2-resident and
// load-issue bound rather than bandwidth bound.
// ---------------------------------------------------------------------------

typedef __attribute__((ext_vector_type(2))) float v2f;
typedef __attribute__((ext_vector_type(8))) float v8f;

#define L0N      3072
#define H1ROWS   30720      // 3072 * 10
#define DIM      128
#define OUTD     256
#define NB       1024       // batch (src/pos)
#define NNEG     512
#define MT       4          // row tiles per wave (M = 64 per wave)

// ---- WMMA helpers ---------------------------------------------------------
// A 16x4 f32 layout: lanes 0-15 hold row M=lane, K={k,k+1}; lanes 16-31 hold
// row M=lane-16, K={k+2,k+3}. B 4x16: VGPR0/1 hold K rows {k,k+1} (lanes 0-15)
// and {k+2,k+3} (lanes 16-31), N = lane%16. C/D: VGPR v = row v + 8*(lane/16).

__device__ __forceinline__ void store_tile(float* __restrict__ C, int ldc,
                                           v8f acc, bool relu) {
    const int lane = threadIdx.x & 31;
    const int n  = lane & 15;
    const int mb = (lane >> 4) * 8;
#pragma unroll
    for (int v = 0; v < 8; ++v) {
        float x = acc[v];
        if (relu) x = fmaxf(x, 0.0f);
        C[(size_t)(mb + v) * ldc + n] = x;
    }
}

// ---- Kernels --------------------------------------------------------------

// dst[i,:] = mean_k src[idx[i*S+k],:]  (idx==nullptr -> dense rows i*S+k)
// C4 = columns/4, one float4 per thread.
__global__ void sage_mean(const float4* __restrict__ src,
                          const int* __restrict__ idx,
                          float4* __restrict__ dst, int rows, int S, int C4) {
    int t = blockIdx.x * blockDim.x + threadIdx.x;
    if (t >= rows * C4) return;
    int i = t / C4;
    int c = t - i * C4;
    float sx = 0.f, sy = 0.f, sz = 0.f, sw = 0.f;
    for (int k = 0; k < S; ++k) {
        int r = idx ? idx[i * S + k] : (i * S + k);
        float4 v = src[(size_t)r * C4 + c];
        sx += v.x; sy += v.y; sz += v.z; sw += v.w;
    }
    float inv = 1.0f / (float)S;
    dst[t] = make_float4(sx * inv, sy * inv, sz * inv, sw * inv);
}

// H[m, 0:128]   = act(Aself[selfIdx[m],:] @ Ws)
// H[m, 128:256] = act(Aneigh[m,:]         @ Wn)
// block = 512 threads = 16 waves; wave w: half = w/8 (self/neigh), ct = w%8.
// Each block covers 64 rows (MT=4 row-tiles per wave); B reused across MT.
__global__ void sage_gemm_concat(const float* __restrict__ Aself,
                                 const int* __restrict__ selfIdx,
                                 const float* __restrict__ Aneigh,
                                 const float* __restrict__ Ws,
                                 const float* __restrict__ Wn,
                                 float* __restrict__ H,
                                 int K, int relu) {
    const int lane = threadIdx.x & 31;
    const int w    = threadIdx.x >> 5;    // 0..15
    const int half = w >> 3;              // 0 = self, 1 = neigh
    const int ct   = w & 7;               // 16-col tile within half
    const int koff = (lane >> 4) << 1;    // 0 or 2
    const int rowb = blockIdx.x * (16 * MT);

    const float* Arow[MT];
#pragma unroll
    for (int j = 0; j < MT; ++j) {
        int m = rowb + j * 16 + (lane & 15);
        if (half == 0) {
            int r = selfIdx ? selfIdx[m] : m;
            Arow[j] = Aself + (size_t)r * K;
        } else {
            Arow[j] = Aneigh + (size_t)m * K;
        }
    }
    const float* Bcol = (half == 0 ? Ws : Wn) + ct * 16 + (lane & 15);

    v8f acc[MT];
#pragma unroll
    for (int j = 0; j < MT; ++j) acc[j] = {};

#pragma unroll 2
    for (int k = 0; k < K; k += 4) {
        v2f b;
        b.x = Bcol[(size_t)(k + koff) * 128];
        b.y = Bcol[(size_t)(k + koff + 1) * 128];
#pragma unroll
        for (int j = 0; j < MT; ++j) {
            v2f a = *reinterpret_cast<const v2f*>(Arow[j] + k + koff);
            acc[j] = __builtin_amdgcn_wmma_f32_16x16x4_f32(
                false, a, false, b, (short)0, acc[j], false, false);
        }
    }

#pragma unroll
    for (int j = 0; j < MT; ++j) {
        float* Cbase = H + (size_t)(rowb + j * 16) * OUTD + half * 128 + ct * 16;
        store_tile(Cbase, OUTD, acc[j], relu != 0);
    }
}

// In-place row L2 normalize of [rows, 256]
__global__ void sage_l2norm(float* __restrict__ out) {
    __shared__ float s[256];
    const int row = blockIdx.x;
    const int t = threadIdx.x;
    float v = out[(size_t)row * OUTD + t];
    s[t] = v * v;
    __syncthreads();
    for (int off = 128; off > 0; off >>= 1) {
        if (t < off) s[t] += s[t + off];
        __syncthreads();
    }
    float scale = rsqrtf(fmaxf(s[0], 1e-12f));
    out[(size_t)row * OUTD + t] = v * scale;
}

// d_out[0:NB*256] = emb[src_idx[i], c]
__global__ void sage_src_out(const float* __restrict__ emb,
                             const int* __restrict__ src_idx,
                             float* __restrict__ outv) {
    int t = blockIdx.x * 256 + threadIdx.x;
    int i = t >> 8;
    int c = t & 255;
    outv[t] = emb[(size_t)src_idx[i] * OUTD + c];
}

// aff[i] = dot(emb[src_idx[i]], emb[pos_idx[i]])
__global__ void sage_aff(const float* __restrict__ emb,
                         const int* __restrict__ src_idx,
                         const int* __restrict__ pos_idx,
                         float* __restrict__ aff) {
    __shared__ float s[256];
    const int i = blockIdx.x;
    const int t = threadIdx.x;
    float p = emb[(size_t)src_idx[i] * OUTD + t] *
              emb[(size_t)pos_idx[i] * OUTD + t];
    s[t] = p;
    __syncthreads();
    for (int off = 128; off > 0; off >>= 1) {
        if (t < off) s[t] += s[t + off];
        __syncthreads();
    }
    if (t == 0) aff[i] = s[0];
}

// negaff[1024,512] = src_emb @ neg_emb^T via WMMA (A*B^T, K=256).
// 16 row-groups (64 rows) x 32 col-tiles = 512 waves; 32 blocks x 16 waves.
// B fragment is one contiguous float2, reused across MT row tiles.
__global__ void sage_negaff(const float* __restrict__ emb,
                            const int* __restrict__ src_idx,
                            const int* __restrict__ neg_idx,
                            float* __restrict__ negaff) {
    const int lane  = threadIdx.x & 31;
    const int w     = threadIdx.x >> 5;
    const int wid   = blockIdx.x * 16 + w;    // 0..511
    const int rg    = wid >> 5;               // 0..15 (64-row group)
    const int ct    = wid & 31;               // 0..31
    const int koff  = (lane >> 4) << 1;

    const float* Arow[MT];
#pragma unroll
    for (int j = 0; j < MT; ++j) {
        int m = rg * (16 * MT) + j * 16 + (lane & 15);
        Arow[j] = emb + (size_t)src_idx[m] * OUTD;
    }
    const float* Brow = emb + (size_t)neg_idx[ct * 16 + (lane & 15)] * OUTD;

    v8f acc[MT];
#pragma unroll
    for (int j = 0; j < MT; ++j) acc[j] = {};

#pragma unroll 2
    for (int k = 0; k < OUTD; k += 4) {
        v2f b = *reinterpret_cast<const v2f*>(Brow + k + koff);
#pragma unroll
        for (int j = 0; j < MT; ++j) {
            v2f a = *reinterpret_cast<const v2f*>(Arow[j] + k + koff);
            acc[j] = __builtin_amdgcn_wmma_f32_16x16x4_f32(
                false, a, false, b, (short)0, acc[j], false, false);
        }
    }

#pragma unroll
    for (int j = 0; j < MT; ++j) {
        store_tile(negaff + (size_t)(rg * 16 * MT + j * 16) * NNEG + ct * 16,
                   NNEG, acc[j], false);
    }
}

__device__ __forceinline__ float softplusf(float x) {
    return fmaxf(x, 0.0f) + log1pf(expf(-fabsf(x)));
}

// Per-row: rank count (neg >= aff, stable-sort tie rule), softplus sums.
__global__ void sage_rowstats(const float* __restrict__ negaff,
                              const float* __restrict__ aff,
                              float* __restrict__ txent,
                              float* __restrict__ nsum,
                              float* __restrict__ mrrv) {
    __shared__ float sf[256];
    __shared__ int   si[256];
    const int i = blockIdx.x;
    const int t = threadIdx.x;
    const float a = aff[i];
    float s = 0.f;
    int c = 0;
    for (int j = t; j < NNEG; j += 256) {
        float na = negaff[(size_t)i * NNEG + j];
        s += softplusf(na);
        c += (na >= a) ? 1 : 0;
    }
    sf[t] = s; si[t] = c;
    __syncthreads();
    for (int off = 128; off > 0; off >>= 1) {
        if (t < off) { sf[t] += sf[t + off]; si[t] += si[t + off]; }
        __syncthreads();
    }
    if (t == 0) {
        nsum[i]  = sf[0];
        mrrv[i]  = 1.0f / (float)(si[0] + 1);
        txent[i] = softplusf(-a);
    }
}

// Deterministic final reduction (fixed-order tree; no float atomics).
__global__ void sage_final(const float* __restrict__ txent,
                           const float* __restrict__ nsum,
                           const float* __restrict__ mrrv,
                           float* __restrict__ outv) {
    __shared__ float sl[256];
    __shared__ float sm[256];
    const int t = threadIdx.x;
    float l = 0.f, m = 0.f;
    for (int i = t; i < NB; i += 256) {
        l += txent[i] + nsum[i];
        m += mrrv[i];
    }
    sl[t] = l; sm[t] = m;
    __syncthreads();
    for (int off = 128; off > 0; off >>= 1) {
        if (t < off) { sl[t] += sl[t + off]; sm[t] += sm[t + off]; }
        __syncthreads();
    }
    if (t == 0) {
        outv[NB * OUTD]     = sl[0] / (float)NB;   // loss
        outv[NB * OUTD + 1] = sm[0] / (float)NB;   // mrr
    }
}

// ---------------------------------------------------------------------------

extern "C" void kernel_launch(void* const* d_in, const int* in_sizes, int n_in,
                              void* d_out, int out_size, void* d_ws, size_t ws_size,
                              hipStream_t stream) {
    (void)in_sizes; (void)n_in; (void)out_size; (void)ws_size;
    // setup_inputs() order:
    // 0 nodes (unused), 1 feat, 2 src_idx, 3 pos_idx, 4 neg_idx,
    // 5 nb0, 6 nb1, 7 nb2, 8 w_self_0, 9 w_neigh_0, 10 w_self_1, 11 w_neigh_1
    const float* feat = (const float*)d_in[1];
    const int* src_idx = (const int*)d_in[2];
    const int* pos_idx = (const int*)d_in[3];
    const int* neg_idx = (const int*)d_in[4];
    const int* nb0 = (const int*)d_in[5];
    const int* nb1 = (const int*)d_in[6];
    const int* nb2 = (const int*)d_in[7];
    const float* Ws0 = (const float*)d_in[8];
    const float* Wn0 = (const float*)d_in[9];
    const float* Ws1 = (const float*)d_in[10];
    const float* Wn1 = (const float*)d_in[11];
    float* out = (float*)d_out;

    // Workspace layout (floats)
    float* ws    = (float*)d_ws;
    float* mean1 = ws;                                  // [30720,128]
    float* h1    = mean1 + (size_t)H1ROWS * DIM;        // [30720,256]
    float* mean0 = h1    + (size_t)H1ROWS * OUTD;       // [3072,128]
    float* h0    = mean0 + (size_t)L0N * DIM;           // [3072,256]
    float* nm    = h0    + (size_t)L0N * OUTD;          // [3072,256]
    float* emb   = nm    + (size_t)L0N * OUTD;          // [3072,256] (pre+post norm)
    float* negaf = emb   + (size_t)L0N * OUTD;          // [1024,512]
    float* aff   = negaf + (size_t)NB * NNEG;           // [1024]
    float* txent = aff   + NB;                          // [1024]
    float* nsum  = txent + NB;                          // [1024]
    float* mrrv  = nsum  + NB;                          // [1024]

    // 1) hop-2 gather-mean: mean1[30720,128] (25-way), mean0[3072,128] (10-way)
    sage_mean<<<dim3((H1ROWS * 32 + 255) / 256), dim3(256), 0, stream>>>(
        (const float4*)feat, nb2, (float4*)mean1, H1ROWS, 25, 32);
    sage_mean<<<dim3((L0N * 32 + 255) / 256), dim3(256), 0, stream>>>(
        (const float4*)feat, nb1, (float4*)mean0, L0N, 10, 32);

    // 2) layer-0 GEMMs (K=128, relu); 64 rows per block
    sage_gemm_concat<<<dim3(H1ROWS / 64), dim3(512), 0, stream>>>(
        feat, nb1, mean1, Ws0, Wn0, h1, DIM, 1);
    sage_gemm_concat<<<dim3(L0N / 64), dim3(512), 0, stream>>>(
        feat, nb0, mean0, Ws0, Wn0, h0, DIM, 1);

    // 3) layer-1: nm = mean10(h1); emb = [h0@Ws1 | nm@Wn1] (K=256, no relu)
    sage_mean<<<dim3((L0N * 64 + 255) / 256), dim3(256), 0, stream>>>(
        (const float4*)h1, nullptr, (float4*)nm, L0N, 10, 64);
    sage_gemm_concat<<<dim3(L0N / 64), dim3(512), 0, stream>>>(
        h0, nullptr, nm, Ws1, Wn1, emb, OUTD, 0);

    // 4) row L2 normalize in place
    sage_l2norm<<<dim3(L0N), dim3(256), 0, stream>>>(emb);

    // 5) outputs: src_emb, aff, neg_aff
    sage_src_out<<<dim3(NB), dim3(256), 0, stream>>>(emb, src_idx, out);
    sage_aff<<<dim3(NB), dim3(256), 0, stream>>>(emb, src_idx, pos_idx, aff);
    sage_negaff<<<dim3(32), dim3(512), 0, stream>>>(emb, src_idx, neg_idx, negaf);

    // 6) rank + softplus stats, then deterministic reduce -> loss, mrr
    sage_rowstats<<<dim3(NB), dim3(256), 0, stream>>>(negaf, aff, txent, nsum, mrrv);
    sage_final<<<dim3(1), dim3(256), 0, stream>>>(txent, nsum, mrrv, out);
}